// MGHD_23845658428098
// MI455X (gfx1250) — compile-verified
//
#include <hip/hip_runtime.h>
#include <hip/hip_bf16.h>

// ---------------------------------------------------------------- constants
#define DIST      21
#define NUM_CHECK 440          // DIST*DIST-1
#define NPG       881          // 2*DIST*DIST-1
#define BG        128
#define NC        (BG * NUM_CHECK)   // 56320 sequence rows
#define TOTAL_N   (BG * NPG)         // 112768 nodes
#define CDIM      128
#define DI        256
#define DS        16
#define DT_RANK   8
#define DCONV     4
#define HID       128
#define N_ITERS   2

typedef float v2f __attribute__((ext_vector_type(2)));
typedef float v8f __attribute__((ext_vector_type(8)));

static __device__ __forceinline__ float silu_f(float x) {
    return x / (1.0f + __expf(-x));
}
static __device__ __forceinline__ float softplus_f(float x) {
    if (x > 20.0f)  return x;
    if (x < -20.0f) return __expf(x);
    return log1pf(__expf(x));
}
static __device__ __forceinline__ v8f wmma_f32x4(v2f a, v2f b, v8f c) {
    // V_WMMA_F32_16X16X4_F32 : D = A(16x4) * B(4x16) + C(16x16), all fp32
    return __builtin_amdgcn_wmma_f32_16x16x4_f32(
        false, a, false, b, (short)0, c, false, false);
}

// =====================================================================
// Full-tile fp32 WMMA GEMM (NO bounds guards):
//   requires M % (16*MT) == 0 and N % (16*NT) == 0, K and K0 % 4 == 0.
// Out = ACT( [A0|A1] * Bp + bias ) (+Res), optional gather/scatter rows.
// Bp packed in k-pairs: Bp[((k>>1)*N + n)*2 + (k&1)].
// One wave owns a (16*MT) x (16*NT) tile; 4 waves / block.
// =====================================================================
template <int NT, int MT, bool GATHER, bool SCATTER, bool RES, bool CONCAT, int ACT>
__global__ __launch_bounds__(128, 1) void gemm_full(
    const float* __restrict__ A0, int lda0, int K0,
    const float* __restrict__ A1, int lda1,
    const float* __restrict__ Bp,
    const float* __restrict__ bias,
    float* __restrict__ Out,
    const float* __restrict__ Resp,
    const int* __restrict__ gatherA,
    const int* __restrict__ scatterO,
    int M, int N, int K)
{
    const int lane = threadIdx.x & 31;
    const int wave = threadIdx.x >> 5;
    const int mi   = lane & 15;
    const int kh   = lane >> 4;

    const int tilesM  = M / (16 * MT);
    const int nStrips = N / (16 * NT);
    const long long tiles = (long long)tilesM * nStrips;
    const long long bid = (long long)blockIdx.x * 4 + wave;
    if (bid >= tiles) return;

    const int strip = (int)(bid % nStrips);
    const int tmi   = (int)(bid / nStrips);
    const int row0  = tmi * 16 * MT;
    const int col0  = strip * 16 * NT;

    int asrc[MT];
#pragma unroll
    for (int mt = 0; mt < MT; ++mt) {
        const int ar = row0 + mt * 16 + mi;
        asrc[mt] = GATHER ? gatherA[ar] : ar;
    }

    v8f acc[MT][NT];
#pragma unroll
    for (int t = 0; t < NT; ++t) {
        const float bv = bias ? bias[col0 + t * 16 + mi] : 0.0f;
#pragma unroll
        for (int mt = 0; mt < MT; ++mt)
#pragma unroll
            for (int r = 0; r < 8; ++r) acc[mt][t][r] = bv;
    }

    const int nseg = CONCAT ? 2 : 1;
#pragma unroll 1
    for (int seg = 0; seg < nseg; ++seg) {
        const float* Aseg = (CONCAT && seg == 1) ? A1 : A0;
        const int    ldas = (CONCAT && seg == 1) ? lda1 : lda0;
        const int    kbeg = (CONCAT && seg == 1) ? K0 : 0;
        const int    kend = (CONCAT && seg == 0) ? K0 : K;

        const float* aBase[MT];
#pragma unroll
        for (int mt = 0; mt < MT; ++mt)
            aBase[mt] = Aseg + (long long)asrc[mt] * ldas + 2 * kh;

        v2f aF[MT], bF[NT];
#pragma unroll
        for (int mt = 0; mt < MT; ++mt)
            aF[mt] = *(const v2f*)(aBase[mt]);
#pragma unroll
        for (int t = 0; t < NT; ++t)
            bF[t] = *(const v2f*)(Bp + ((long long)((kbeg + 2 * kh) >> 1) * N +
                                        (col0 + t * 16 + mi)) * 2);

        for (int k0 = kbeg; k0 < kend; k0 += 4) {
            v2f aN[MT], bN[NT];
            const bool more = (k0 + 4 < kend);
            if (more) {
#pragma unroll
                for (int mt = 0; mt < MT; ++mt)
                    aN[mt] = *(const v2f*)(aBase[mt] + (k0 + 4 - kbeg));
#pragma unroll
                for (int t = 0; t < NT; ++t)
                    bN[t] = *(const v2f*)(Bp + ((long long)((k0 + 4 + 2 * kh) >> 1) * N +
                                                (col0 + t * 16 + mi)) * 2);
            }
#pragma unroll
            for (int mt = 0; mt < MT; ++mt)
#pragma unroll
                for (int t = 0; t < NT; ++t)
                    acc[mt][t] = wmma_f32x4(aF[mt], bF[t], acc[mt][t]);
            if (more) {
#pragma unroll
                for (int mt = 0; mt < MT; ++mt) aF[mt] = aN[mt];
#pragma unroll
                for (int t = 0; t < NT; ++t)   bF[t] = bN[t];
            }
        }
    }

    // ---------------- epilogue ----------------------------------------
#pragma unroll
    for (int mt = 0; mt < MT; ++mt) {
        int drow[8];
#pragma unroll
        for (int r = 0; r < 8; ++r) {
            const int row = row0 + mt * 16 + r + 8 * kh;
            drow[r] = SCATTER ? scatterO[row] : row;
        }
#pragma unroll
        for (int t = 0; t < NT; ++t) {
            const int col = col0 + t * 16 + mi;
#pragma unroll
            for (int r = 0; r < 8; ++r) {
                float v = acc[mt][t][r];
                if (ACT == 1)      v = silu_f(v);
                else if (ACT == 2) v = softplus_f(v);
                if (RES) v += Resp[(long long)(row0 + mt * 16 + r + 8 * kh) * N + col];
                if (SCATTER) atomicAdd(&Out[(long long)drow[r] * N + col], v);
                else         Out[(long long)drow[r] * N + col] = v;
            }
        }
    }
}

// =====================================================================
// Guarded edge GEMM (only used for x_proj, N=40): simple, small.
// =====================================================================
template <int NT>
__global__ void gemm_edge(
    const float* __restrict__ A0, int lda0,
    const float* __restrict__ Bp,
    float* __restrict__ Out,
    int M, int N, int K)
{
    const int lane = threadIdx.x & 31;
    const int wave = threadIdx.x >> 5;
    const int mi   = lane & 15;
    const int kh   = lane >> 4;

    const int tilesM  = (M + 15) >> 4;
    const long long bid = (long long)blockIdx.x * 4 + wave;
    if (bid >= tilesM) return;
    const int row0 = (int)bid * 16;

    const bool rok  = (row0 + mi < M);
    const int  asrc = rok ? (row0 + mi) : 0;

    v8f acc[NT];
#pragma unroll
    for (int t = 0; t < NT; ++t)
#pragma unroll
        for (int r = 0; r < 8; ++r) acc[t][r] = 0.0f;

    for (int k0 = 0; k0 < K; k0 += 4) {
        const int ka = k0 + 2 * kh;
        v2f a;
        if (rok) a = *(const v2f*)(A0 + (long long)asrc * lda0 + ka);
        else { a.x = 0.0f; a.y = 0.0f; }
#pragma unroll
        for (int t = 0; t < NT; ++t) {
            const int col = t * 16 + mi;
            v2f b;
            if (col < N) b = *(const v2f*)(Bp + ((long long)(ka >> 1) * N + col) * 2);
            else { b.x = 0.0f; b.y = 0.0f; }
            acc[t] = wmma_f32x4(a, b, acc[t]);
        }
    }
#pragma unroll
    for (int t = 0; t < NT; ++t) {
        const int col = t * 16 + mi;
        if (col >= N) continue;
#pragma unroll
        for (int r = 0; r < 8; ++r) {
            const int row = row0 + r + 8 * kh;
            if (row >= M) continue;
            Out[(long long)row * N + col] = acc[t][r];
        }
    }
}

// pack W[KxN] row-major into k-pair order: Wp[((k>>1)*N + n)*2 + (k&1)]
__global__ void pack_pairs_kernel(const float* __restrict__ W,
                                  float* __restrict__ Wp, int K, int N) {
    const int i = blockIdx.x * blockDim.x + threadIdx.x;
    if (i >= K * N) return;
    const int k = i / N, n = i - k * N;
    Wp[((long long)(k >> 1) * N + n) * 2 + (k & 1)] = W[i];
}

// ================================================================ misc VALU
__global__ void fill_zero_kernel(float* __restrict__ p, long long n) {
    long long i = (long long)blockIdx.x * blockDim.x + threadIdx.x;
    const long long stride = (long long)gridDim.x * blockDim.x;
    for (; i < n; i += stride) p[i] = 0.0f;
}

__global__ void embed_kernel(const float* __restrict__ node_inputs,
                             const float* __restrict__ emb_w,
                             const float* __restrict__ emb_b,
                             float* __restrict__ emb) {
    const int tid = blockIdx.x * blockDim.x + threadIdx.x;
    if (tid >= NC * CDIM) return;
    const int c = tid & (CDIM - 1);
    const int r = tid >> 7;
    const int g = r / NUM_CHECK, j = r - g * NUM_CHECK;
    const int node = g * NPG + j;
    float acc = emb_b[c];
#pragma unroll
    for (int k = 0; k < 4; ++k)
        acc += node_inputs[node * 4 + k] * emb_w[k * CDIM + c];
    emb[(long long)r * CDIM + c] = acc;
}

__global__ void conv_silu_kernel(const float* __restrict__ xz,   // [NC,512]
                                 const float* __restrict__ conv_w,
                                 const float* __restrict__ conv_b,
                                 float* __restrict__ u) {        // [NC,256]
    const int tid = blockIdx.x * blockDim.x + threadIdx.x;
    if (tid >= NC * DI) return;
    const int d = tid & (DI - 1);
    const int r = tid >> 8;
    const int b = r / NUM_CHECK, l = r - b * NUM_CHECK;
    float acc = conv_b[d];
#pragma unroll
    for (int k = 0; k < DCONV; ++k) {
        const int ll = l + k - (DCONV - 1);
        if (ll >= 0)
            acc += conv_w[d * DCONV + k] *
                   xz[((long long)(b * NUM_CHECK + ll)) * (2 * DI) + d];
    }
    u[(long long)r * DI + d] = silu_f(acc);
}

// selective scan fused with  y = (y + u*D) * silu(zg); overwrites delta
__global__ void scan_kernel(const float* __restrict__ u,      // [NC,256]
                            float* __restrict__ delta,        // [NC,256] in/out
                            const float* __restrict__ xdb,    // [NC,40]
                            const float* __restrict__ xz,     // [NC,512]
                            const float* __restrict__ A_log,  // [256,16]
                            const float* __restrict__ Dp) {   // [256]
    const int b = blockIdx.x;
    const int d = threadIdx.x;
    __shared__ float sBC[2 * DS];

    float As[DS];
#pragma unroll
    for (int s = 0; s < DS; ++s) As[s] = -__expf(A_log[d * DS + s]);
    float h[DS];
#pragma unroll
    for (int s = 0; s < DS; ++s) h[s] = 0.0f;
    const float Dd = Dp[d];

    for (int l = 0; l < NUM_CHECK; ++l) {
        const long long row = (long long)b * NUM_CHECK + l;
        if (d < 2 * DS) sBC[d] = xdb[row * (DT_RANK + 2 * DS) + DT_RANK + d];
        __syncthreads();
        const float dlt = delta[row * DI + d];
        const float ut  = u[row * DI + d];
        const float zg  = xz[row * (2 * DI) + DI + d];
        const float du  = dlt * ut;
        float y = 0.0f;
#pragma unroll
        for (int s = 0; s < DS; ++s) {
            const float dA = __expf(dlt * As[s]);
            h[s] = dA * h[s] + du * sBC[s];
            y += h[s] * sBC[DS + s];
        }
        y = (y + ut * Dd) * silu_f(zg);
        delta[row * DI + d] = y;
        __syncthreads();
    }
}

__global__ void se_mean_kernel(const float* __restrict__ seq,
                               float* __restrict__ smean) {
    const int tid = blockIdx.x * blockDim.x + threadIdx.x;
    if (tid >= BG * CDIM) return;
    const int c = tid & (CDIM - 1);
    const int g = tid >> 7;
    float acc = 0.0f;
    for (int l = 0; l < NUM_CHECK; ++l)
        acc += seq[((long long)g * NUM_CHECK + l) * CDIM + c];
    smean[g * CDIM + c] = acc * (1.0f / NUM_CHECK);
}

__global__ void se_gate_kernel(const float* __restrict__ smean,
                               const float* __restrict__ w1, const float* __restrict__ b1,
                               const float* __restrict__ w2, const float* __restrict__ b2,
                               float* __restrict__ gate) {
    const int g = blockIdx.x, c = threadIdx.x;
    __shared__ float ss[CDIM];
    __shared__ float tt[CDIM / 4];
    ss[c] = smean[g * CDIM + c];
    __syncthreads();
    if (c < CDIM / 4) {
        float acc = b1[c];
        for (int j = 0; j < CDIM; ++j) acc += ss[j] * w1[j * (CDIM / 4) + c];
        tt[c] = silu_f(acc);
    }
    __syncthreads();
    float acc = b2[c];
    for (int j = 0; j < CDIM / 4; ++j) acc += tt[j] * w2[j * CDIM + c];
    gate[g * CDIM + c] = 1.0f / (1.0f + __expf(-acc));
}

__global__ void proj_scatter_kernel(const float* __restrict__ seq,
                                    const float* __restrict__ gate,
                                    const float* __restrict__ proj_w,
                                    const float* __restrict__ proj_b,
                                    float* __restrict__ gnn_in) {
    const int tid = blockIdx.x * blockDim.x + threadIdx.x;
    if (tid >= NC * 4) return;
    const int o = tid & 3;
    const int r = tid >> 2;
    const int g = r / NUM_CHECK, j = r - g * NUM_CHECK;
    float acc = proj_b[o];
    const float* sp = seq + (long long)r * CDIM;
    const float* gp = gate + g * CDIM;
    for (int c = 0; c < CDIM; ++c)
        acc += sp[c] * gp[c] * proj_w[c * 4 + o];
    gnn_in[((long long)g * NPG + j) * 4 + o] = acc;
}

__global__ void gnn_in_kernel(const float* __restrict__ gnn_in,
                              const float* __restrict__ w, const float* __restrict__ b,
                              float* __restrict__ h) {
    const int tid = blockIdx.x * blockDim.x + threadIdx.x;
    if (tid >= TOTAL_N * HID) return;
    const int c = tid & (HID - 1);
    const int n = tid >> 7;
    float acc = b[c];
#pragma unroll
    for (int k = 0; k < 4; ++k)
        acc += gnn_in[(long long)n * 4 + k] * w[k * HID + c];
    h[(long long)n * HID + c] = silu_f(acc);
}

__global__ void gnn_out_kernel(const float* __restrict__ h,
                               const float* __restrict__ w, const float* __restrict__ b,
                               float* __restrict__ out) {
    const int tid = blockIdx.x * blockDim.x + threadIdx.x;
    if (tid >= TOTAL_N * 2) return;
    const int o = tid & 1;
    const int n = tid >> 1;
    float acc = b[o];
    const float* hp = h + (long long)n * HID;
    for (int c = 0; c < HID; ++c) acc += hp[c] * w[c * 2 + o];
    out[(long long)n * 2 + o] = acc;
}

// ================================================================= launch
static inline int full_blocks(int M, int N, int NT, int MT) {
    const long long tiles = ((long long)M / (16 * MT)) * (N / (16 * NT));
    return (int)((tiles + 3) / 4);
}

extern "C" void kernel_launch(void* const* d_in, const int* in_sizes, int n_in,
                              void* d_out, int out_size, void* d_ws, size_t ws_size,
                              hipStream_t stream) {
    const float* node_inputs = (const float*)d_in[0];
    const int*   src_ids     = (const int*)d_in[1];
    const int*   dst_ids     = (const int*)d_in[2];
    const int    E           = in_sizes[1];

    const float* emb_w      = (const float*)d_in[3];
    const float* emb_b      = (const float*)d_in[4];
    const float* in_proj_w  = (const float*)d_in[5];
    const float* in_proj_b  = (const float*)d_in[6];
    const float* conv_w     = (const float*)d_in[7];
    const float* conv_b     = (const float*)d_in[8];
    const float* x_proj_w   = (const float*)d_in[9];
    const float* dt_proj_w  = (const float*)d_in[10];
    const float* dt_proj_b  = (const float*)d_in[11];
    const float* A_log      = (const float*)d_in[12];
    const float* Dp         = (const float*)d_in[13];
    const float* out_proj_w = (const float*)d_in[14];
    const float* out_proj_b = (const float*)d_in[15];
    const float* se1_w      = (const float*)d_in[16];
    const float* se1_b      = (const float*)d_in[17];
    const float* se2_w      = (const float*)d_in[18];
    const float* se2_b      = (const float*)d_in[19];
    const float* proj_w     = (const float*)d_in[20];
    const float* proj_b     = (const float*)d_in[21];
    const float* gnn_in_w   = (const float*)d_in[22];
    const float* gnn_in_b   = (const float*)d_in[23];
    const float* gnn_out_w  = (const float*)d_in[24];
    const float* gnn_out_b  = (const float*)d_in[25];
    const float* msg_w[N_ITERS] = {(const float*)d_in[26], (const float*)d_in[30]};
    const float* msg_b[N_ITERS] = {(const float*)d_in[27], (const float*)d_in[31]};
    const float* upd_w[N_ITERS] = {(const float*)d_in[28], (const float*)d_in[32]};
    const float* upd_b[N_ITERS] = {(const float*)d_in[29], (const float*)d_in[33]};

    float* ws = (float*)d_ws;
    // ---- workspace arena (floats) -------------------------------------
    const size_t oXZ    = 0;                                  // [NC,512]
    const size_t oU     = oXZ    + (size_t)NC * 512;          // [NC,256]
    const size_t oDelta = oU     + (size_t)NC * DI;           // [NC,256]
    const size_t oSeq   = oDelta + (size_t)NC * DI;           // [NC,128] (emb too)
    const size_t oXdb   = oSeq   + (size_t)NC * CDIM;         // [NC,40]
    const size_t oSmean = oXdb   + (size_t)NC * 40;
    const size_t oGate  = oSmean + (size_t)BG * CDIM;
    const size_t oGnnIn = oGate  + (size_t)BG * CDIM;         // [TOTAL,4]
    const size_t oPk    = oGnnIn + (size_t)TOTAL_N * 4;       // packed weights
    float* xz    = ws + oXZ;
    float* u     = ws + oU;
    float* delta = ws + oDelta;
    float* seq   = ws + oSeq;
    float* emb   = ws + oSeq;
    float* xdb   = ws + oXdb;
    float* smean = ws + oSmean;
    float* gate  = ws + oGate;
    float* gnnin = ws + oGnnIn;
    // packed weights
    float* pk_in_proj  = ws + oPk;                         // 128*512
    float* pk_x_proj   = pk_in_proj  + (size_t)CDIM * 512; // 256*40
    float* pk_dt_proj  = pk_x_proj   + (size_t)DI * 40;    // 8*256
    float* pk_out_proj = pk_dt_proj  + (size_t)DT_RANK * DI;
    float* pk_msg[N_ITERS];
    float* pk_upd[N_ITERS];
    pk_msg[0] = pk_out_proj + (size_t)DI * CDIM;
    pk_upd[0] = pk_msg[0] + (size_t)HID * HID;
    pk_msg[1] = pk_upd[0] + (size_t)(2 * HID) * HID;
    pk_upd[1] = pk_msg[1] + (size_t)HID * HID;
    // GNN phase reuses the (dead) Mamba activation region
    float* hA  = ws;
    float* hB  = ws + (size_t)TOTAL_N * HID;
    float* agg = ws + (size_t)2 * TOTAL_N * HID;

    float* out = (float*)d_out;

    // ---- pack weights into k-pair order -------------------------------
    auto pack = [&](const float* W, float* Wp, int K, int N) {
        pack_pairs_kernel<<<(K * N + 255) / 256, 256, 0, stream>>>(W, Wp, K, N);
    };
    pack(in_proj_w,  pk_in_proj,  CDIM, 512);
    pack(x_proj_w,   pk_x_proj,   DI, 40);
    pack(dt_proj_w,  pk_dt_proj,  DT_RANK, DI);
    pack(out_proj_w, pk_out_proj, DI, CDIM);
    for (int i = 0; i < N_ITERS; ++i) {
        pack(msg_w[i], pk_msg[i], HID, HID);
        pack(upd_w[i], pk_upd[i], 2 * HID, HID);
    }

    // ---- encoder ------------------------------------------------------
    embed_kernel<<<(NC * CDIM + 255) / 256, 256, 0, stream>>>(
        node_inputs, emb_w, emb_b, emb);

    // in_proj: [NC,128] x [128,512]
    gemm_full<8, 2, false, false, false, false, 0>
        <<<full_blocks(NC, 512, 8, 2), 128, 0, stream>>>(
        emb, CDIM, CDIM, nullptr, 0, pk_in_proj, in_proj_b,
        xz, nullptr, nullptr, nullptr, NC, 512, CDIM);

    conv_silu_kernel<<<(NC * DI + 255) / 256, 256, 0, stream>>>(
        xz, conv_w, conv_b, u);

    // x_proj: [NC,256] x [256,40]  (edge kernel, N=40)
    gemm_edge<3><<<(((NC + 15) / 16) + 3) / 4, 128, 0, stream>>>(
        u, DI, pk_x_proj, xdb, NC, 40, DI);

    // dt_proj + softplus: [NC,8] x [8,256] (A stride 40)
    gemm_full<8, 2, false, false, false, false, 2>
        <<<full_blocks(NC, DI, 8, 2), 128, 0, stream>>>(
        xdb, DT_RANK + 2 * DS, DT_RANK, nullptr, 0, pk_dt_proj, dt_proj_b,
        delta, nullptr, nullptr, nullptr, NC, DI, DT_RANK);

    scan_kernel<<<BG, DI, 0, stream>>>(u, delta, xdb, xz, A_log, Dp);

    // out_proj: [NC,256] x [256,128]
    gemm_full<8, 2, false, false, false, false, 0>
        <<<full_blocks(NC, CDIM, 8, 2), 128, 0, stream>>>(
        delta, DI, DI, nullptr, 0, pk_out_proj, out_proj_b,
        seq, nullptr, nullptr, nullptr, NC, CDIM, DI);

    se_mean_kernel<<<(BG * CDIM + 255) / 256, 256, 0, stream>>>(seq, smean);
    se_gate_kernel<<<BG, CDIM, 0, stream>>>(smean, se1_w, se1_b, se2_w, se2_b, gate);

    fill_zero_kernel<<<2048, 256, 0, stream>>>(gnnin, (long long)TOTAL_N * 4);
    proj_scatter_kernel<<<(NC * 4 + 255) / 256, 256, 0, stream>>>(
        seq, gate, proj_w, proj_b, gnnin);

    // ---- GNN decoder --------------------------------------------------
    gnn_in_kernel<<<(TOTAL_N * HID + 255) / 256, 256, 0, stream>>>(
        gnnin, gnn_in_w, gnn_in_b, hA);

    float* hcur = hA;
    float* hnxt = hB;
    for (int it = 0; it < N_ITERS; ++it) {
        fill_zero_kernel<<<4096, 256, 0, stream>>>(agg, (long long)TOTAL_N * HID);
        // m = silu(h[src] @ msg_w + msg_b); agg[dst] += m  (gather + atomic scatter)
        gemm_full<8, 2, true, true, false, false, 1>
            <<<full_blocks(E, HID, 8, 2), 128, 0, stream>>>(
            hcur, HID, HID, nullptr, 0, pk_msg[it], msg_b[it],
            agg, nullptr, src_ids, dst_ids, E, HID, HID);
        // h' = h + silu([h|agg] @ upd_w + upd_b)
        gemm_full<8, 2, false, false, true, true, 1>
            <<<full_blocks(TOTAL_N, HID, 8, 2), 128, 0, stream>>>(
            hcur, HID, HID, agg, HID, pk_upd[it], upd_b[it],
            hnxt, hcur, nullptr, nullptr, TOTAL_N, HID, 2 * HID);
        float* tmp = hcur; hcur = hnxt; hnxt = tmp;
    }

    gnn_out_kernel<<<(TOTAL_N * 2 + 255) / 256, 256, 0, stream>>>(
        hcur, gnn_out_w, gnn_out_b, out);

    (void)n_in; (void)out_size; (void)ws_size;
}